// AsyncCrossModalConsistencyLoss_38809324487140
// MI455X (gfx1250) — compile-verified
//
#include <hip/hip_runtime.h>
#include <hip/hip_bf16.h>
#include <math.h>

typedef __attribute__((ext_vector_type(2))) float v2f;
typedef __attribute__((ext_vector_type(8))) float v8f;

#define B_ 8
#define S_ 4096
#define D_ 512
#define ROWS_PER_BLOCK 64
#define ROWS_PER_WAVE 8
#define BLOCKS_PER_B (S_ / ROWS_PER_BLOCK) /* 64 */

// ---------------------------------------------------------------------------
// Workspace layout (floats):
//   [0 .. 4095]      gSumV : per-batch sum of normalized visual rows  [8][512]
//   [4096 .. 8191]   gSumA : per-batch sum of normalized audio rows   [8][512]
//   [8192 .. 8199]   gSync : per-batch sum of <v_hat, a_hat>          [8]
// ---------------------------------------------------------------------------

__global__ void zero_ws_kernel(float* __restrict__ ws, int n) {
  int i = blockIdx.x * blockDim.x + threadIdx.x;
  if (i < n) ws[i] = 0.0f;
}

// Streaming single-pass reduction. Each wave32 owns one row at a time:
// lane l loads 16 floats per tensor as 4 fully-coalesced float4s
// (d = j*128 + l*4), computes vv/aa/va partials, butterfly-reduces across
// the wave, then accumulates the normalized row into register accumulators.
__global__ __launch_bounds__(256) void rowreduce_kernel(
    const float* __restrict__ V, const float* __restrict__ A,
    float* __restrict__ gV, float* __restrict__ gA, float* __restrict__ gS) {
  __shared__ float sV[D_];
  __shared__ float sA[D_];
  __shared__ float sSync;

  const int tid = threadIdx.x;
  sV[tid] = 0.0f; sV[tid + 256] = 0.0f;
  sA[tid] = 0.0f; sA[tid + 256] = 0.0f;
  if (tid == 0) sSync = 0.0f;
  __syncthreads();

  const int b  = blockIdx.x / BLOCKS_PER_B;
  const int s0 = (blockIdx.x % BLOCKS_PER_B) * ROWS_PER_BLOCK;
  const int w  = tid >> 5;
  const int l  = tid & 31;

  float4 accv[4], acca[4];
#pragma unroll
  for (int j = 0; j < 4; ++j) {
    accv[j] = make_float4(0.f, 0.f, 0.f, 0.f);
    acca[j] = make_float4(0.f, 0.f, 0.f, 0.f);
  }
  float syncAcc = 0.0f;

  for (int i = 0; i < ROWS_PER_WAVE; ++i) {
    const int s = s0 + w * ROWS_PER_WAVE + i;
    const float* vrow = V + ((size_t)(b * S_ + s)) * D_;
    const float* arow = A + ((size_t)(b * S_ + s)) * D_;

    if (i + 1 < ROWS_PER_WAVE) {  // gfx1250 global_prefetch_b8 of next row
      __builtin_prefetch(vrow + D_ + l * 4, 0, 1);
      __builtin_prefetch(arow + D_ + l * 4, 0, 1);
    }

    float4 v[4], a[4];
#pragma unroll
    for (int j = 0; j < 4; ++j) {
      v[j] = *reinterpret_cast<const float4*>(vrow + j * 128 + l * 4);
      a[j] = *reinterpret_cast<const float4*>(arow + j * 128 + l * 4);
    }

    float vv = 0.f, aa = 0.f, va = 0.f;
#pragma unroll
    for (int j = 0; j < 4; ++j) {
      vv += v[j].x * v[j].x + v[j].y * v[j].y + v[j].z * v[j].z + v[j].w * v[j].w;
      aa += a[j].x * a[j].x + a[j].y * a[j].y + a[j].z * a[j].z + a[j].w * a[j].w;
      va += v[j].x * a[j].x + v[j].y * a[j].y + v[j].z * a[j].z + v[j].w * a[j].w;
    }

#pragma unroll
    for (int off = 16; off > 0; off >>= 1) {
      vv += __shfl_xor(vv, off, 32);
      aa += __shfl_xor(aa, off, 32);
      va += __shfl_xor(va, off, 32);
    }

    const float inv_v = 1.0f / fmaxf(sqrtf(vv), 1e-12f);
    const float inv_a = 1.0f / fmaxf(sqrtf(aa), 1e-12f);
    syncAcc += va * inv_v * inv_a;

#pragma unroll
    for (int j = 0; j < 4; ++j) {
      accv[j].x += v[j].x * inv_v; accv[j].y += v[j].y * inv_v;
      accv[j].z += v[j].z * inv_v; accv[j].w += v[j].w * inv_v;
      acca[j].x += a[j].x * inv_a; acca[j].y += a[j].y * inv_a;
      acca[j].z += a[j].z * inv_a; acca[j].w += a[j].w * inv_a;
    }
  }

  // Combine the 8 waves of this block in LDS (ds_add_f32 atomics).
#pragma unroll
  for (int j = 0; j < 4; ++j) {
    const int base = j * 128 + l * 4;
    atomicAdd(&sV[base + 0], accv[j].x); atomicAdd(&sV[base + 1], accv[j].y);
    atomicAdd(&sV[base + 2], accv[j].z); atomicAdd(&sV[base + 3], accv[j].w);
    atomicAdd(&sA[base + 0], acca[j].x); atomicAdd(&sA[base + 1], acca[j].y);
    atomicAdd(&sA[base + 2], acca[j].z); atomicAdd(&sA[base + 3], acca[j].w);
  }
  if (l == 0) atomicAdd(&sSync, syncAcc);
  __syncthreads();

  // One global f32 atomic per (b,d) per block: 64 adds per counter total.
  atomicAdd(&gV[b * D_ + tid],       sV[tid]);
  atomicAdd(&gV[b * D_ + tid + 256], sV[tid + 256]);
  atomicAdd(&gA[b * D_ + tid],       sA[tid]);
  atomicAdd(&gA[b * D_ + tid + 256], sA[tid + 256]);
  if (tid == 0) atomicAdd(&gS[b], sSync);
}

// Epilogue: total_sum[b] = <gSumV[b,:], gSumA[b,:]> for b=0..7 computed on
// the matrix unit via chained v_wmma_f32_16x16x4_f32 accumulation.
// M-slot m maps to batch m&7 (A operand), N-slot n to batch n&7 (B operand);
// the diagonal C[b][b] is the desired dot product. One full wave, EXEC=all-1s.
__global__ __launch_bounds__(32) void finalize_kernel(
    const float* __restrict__ ws, const int* __restrict__ targets,
    float* __restrict__ out) {
  const float* gV = ws;
  const float* gA = ws + B_ * D_;
  const float* gS = ws + 2 * B_ * D_;

  const int l    = threadIdx.x;
  const int half = l >> 4;         // 0 -> K{0,1}, 1 -> K{2,3}
  const int brow = (l & 15) & 7;   // batch for this M/N slot

  v8f c = {0.f, 0.f, 0.f, 0.f, 0.f, 0.f, 0.f, 0.f};
  for (int t = 0; t < 128; ++t) {
    const int d = t * 4 + half * 2;
    v2f av = *reinterpret_cast<const v2f*>(gV + brow * D_ + d);  // A: 16x4 f32
    v2f bv = *reinterpret_cast<const v2f*>(gA + brow * D_ + d);  // B: 4x16 f32
    c = __builtin_amdgcn_wmma_f32_16x16x4_f32(
        /*neg_a=*/false, av, /*neg_b=*/false, bv,
        /*c_mod=*/(short)0, c, /*reuse_a=*/false, /*reuse_b=*/false);
  }

  // Dump the 16x16 C tile to LDS per the CDNA5 C/D layout, read the diagonal.
  __shared__ float ctile[16][16];
  __shared__ float lossv[8];
  const int n     = l & 15;
  const int mbase = (l >> 4) * 8;
#pragma unroll
  for (int r = 0; r < 8; ++r) ctile[mbase + r][n] = c[r];
  __syncthreads();

  if (l < 8) {
    const float total     = ctile[l][l];          // sum over full S x S sims
    const float diag      = gS[l];                // sum of diagonal sims
    const float sync_mean = diag * (1.0f / (float)S_);
    const float async_mean =
        (total - diag) / ((float)S_ * (float)(S_ - 1) + 1e-8f);
    const float wt   = (float)targets[l];
    const float loss = wt * fmaxf(async_mean - sync_mean + 0.5f, 0.0f) +
                       (1.0f - wt) * fmaxf(sync_mean - async_mean + 0.05f, 0.0f);
    lossv[l] = loss;
  }
  __syncthreads();

  if (l == 0) {
    float s = 0.0f;
#pragma unroll
    for (int i = 0; i < 8; ++i) s += lossv[i];
    out[0] = s * (1.0f / (float)B_);
  }
}

extern "C" void kernel_launch(void* const* d_in, const int* in_sizes, int n_in,
                              void* d_out, int out_size, void* d_ws, size_t ws_size,
                              hipStream_t stream) {
  (void)in_sizes; (void)n_in; (void)out_size; (void)ws_size;
  const float* V   = (const float*)d_in[0];  // visual_features [8,4096,512] f32
  const float* A   = (const float*)d_in[1];  // audio_features  [8,4096,512] f32
  const int*   tgt = (const int*)d_in[2];    // targets [8] i32
  float* ws  = (float*)d_ws;
  float* out = (float*)d_out;

  const int nz = 2 * B_ * D_ + B_;  // 8200 floats of workspace accumulators
  zero_ws_kernel<<<(nz + 255) / 256, 256, 0, stream>>>(ws, nz);

  rowreduce_kernel<<<B_ * BLOCKS_PER_B, 256, 0, stream>>>(
      V, A, ws, ws + B_ * D_, ws + 2 * B_ * D_);

  finalize_kernel<<<1, 32, 0, stream>>>(ws, tgt, out);
}